// Raymarcher_37864431682184
// MI455X (gfx1250) — compile-verified
//
#include <hip/hip_runtime.h>
#include <hip/hip_bf16.h>

// ---------------------------------------------------------------------------
// Raymarcher for MI455X (gfx1250, wave32).
//   B=1, H=W=256, K=64 prims, M=16 voxel grid, 64 steps, DT=1/256.
// Dense per-(ray,step,prim) affine transform is done on the matrix cores via
// V_WMMA_F32_16X16X4_F32 (16 rays x [x,y,z,1] x 4x192 fused transform).
// Sparse trilinear gather reads a voxel-major float4 repack of the template.
// ---------------------------------------------------------------------------

#define H_DIM 256
#define W_DIM 256
#define NRAYS (H_DIM * W_DIM)      // 65536
#define KPRIM 64
#define MV 16
#define NSTEPS 64
#define DT (1.0f / 256.0f)
#define INV_VOLR (1.0f / 256.0f)

typedef __attribute__((ext_vector_type(2))) float v2f;
typedef __attribute__((ext_vector_type(8))) float v8f;

// --------------------------- template repack -------------------------------
// primrgba [K][4][16^3] (channel-major)  ->  tex4 [K][16^3][4] (voxel-major)
__global__ void repack_kernel(const float* __restrict__ primrgba,
                              float4* __restrict__ tex4) {
    int idx = blockIdx.x * blockDim.x + threadIdx.x;   // 0 .. K*4096-1
    if (idx >= KPRIM * 4096) return;
    int k = idx >> 12;
    int v = idx & 4095;
    const float* src = primrgba + k * 16384;
    float4 o;
    o.x = src[0 * 4096 + v];
    o.y = src[1 * 4096 + v];
    o.z = src[2 * 4096 + v];
    o.w = src[3 * 4096 + v];
    tex4[idx] = o;
}

// ------------------------------ raymarch -----------------------------------
// Block = 128 threads = 4 waves. Each wave marches 16 rays.
// Lane L and L+16 carry the SAME ray (L&15) so the WMMA A-matrix layout
// (lanes 0-15: {x,y}=K0,K1 ; lanes 16-31: {z,1}=K2,K3) needs no shuffles.
__global__ __launch_bounds__(128)
void march_kernel(const float* __restrict__ raypos,
                  const float* __restrict__ raydir,
                  const float* __restrict__ tminmax,
                  const float* __restrict__ primpos,
                  const float* __restrict__ primrot,
                  const float* __restrict__ primscale,
                  const float* __restrict__ primrgba,
                  const float4* __restrict__ tex4,   // may be null
                  float* __restrict__ out) {
    __shared__ float smem[4 * 16 * 192];   // 12 KB per wave: locals[16][192]

    const int lane   = threadIdx.x & 31;
    const int wave   = threadIdx.x >> 5;
    const int laneHi = lane >> 4;          // 0: lanes 0-15, 1: lanes 16-31
    const int row    = lane & 15;          // ray-in-tile this lane owns
    const int r0     = (blockIdx.x * 4 + wave) * 16;  // first global ray
    const int rr     = r0 + row;           // this lane's global ray id

    float* w_lds = smem + wave * (16 * 192);

    // ---- per-ray state (duplicated across the two lane halves) ----
    const float ox = raypos[rr * 3 + 0];
    const float oy = raypos[rr * 3 + 1];
    const float oz = raypos[rr * 3 + 2];
    const float dx = raydir[rr * 3 + 0];
    const float dy = raydir[rr * 3 + 1];
    const float dz = raydir[rr * 3 + 2];
    const float tmin = tminmax[rr * 2 + 0];
    const float tmax = tminmax[rr * 2 + 1];

    // ---- build the 12 B-tiles of the fused 4x192 transform ----
    // Column c = 3*k + j:  local_j = s*(R00j*x + R1j*y + R2j*z) - s*(R^T p)_j
    // B tile layout (4x16 f32): VGPR0 = {K0 | K2}, VGPR1 = {K1 | K3}.
    v2f bt[12];
#pragma unroll
    for (int T = 0; T < 12; ++T) {
        int c = T * 16 + row;
        int k = c / 3, j = c - 3 * k;
        float s  = primscale[k * 3 + j];
        float m0 = primrot[k * 9 + 0 * 3 + j] * s;
        float m1 = primrot[k * 9 + 1 * 3 + j] * s;
        float m2 = primrot[k * 9 + 2 * 3 + j] * s;
        float px = primpos[k * 3 + 0] * INV_VOLR;
        float py = primpos[k * 3 + 1] * INV_VOLR;
        float pz = primpos[k * 3 + 2] * INV_VOLR;
        float b3 = -(m0 * px + m1 * py + m2 * pz);
        bt[T].x = laneHi ? m2 : m0;   // K=2 : K=0
        bt[T].y = laneHi ? b3 : m1;   // K=3 : K=1
    }

    float rgbx = 0.f, rgby = 0.f, rgbz = 0.f, alpha = 0.f;
    const v8f czero = {};

#pragma unroll 1
    for (int i = 0; i < NSTEPS; ++i) {
        const float t  = tmin + ((float)i + 0.5f) * DT;
        const float px = ox + t * dx;
        const float py = oy + t * dy;
        const float pz = oz + t * dz;

        // A-matrix (16x4 f32, 2 VGPRs): lanes<16 {x,y}, lanes>=16 {z,1}
        v2f a;
        a.x = laneHi ? pz : px;
        a.y = laneHi ? 1.0f : py;

        // 12 WMMAs cover all 192 (prim,coord) columns; scatter D to LDS.
#pragma unroll
        for (int T = 0; T < 12; ++T) {
            v8f d = __builtin_amdgcn_wmma_f32_16x16x4_f32(
                false, a, false, bt[T], (short)0, czero, false, false);
#pragma unroll
            for (int v = 0; v < 8; ++v) {
                // D layout: VGPR v, lane n -> (M = v + 8*(n>=16), N = n&15)
                w_lds[(v + (laneHi << 3)) * 192 + T * 16 + row] = d[v];
            }
        }
        __syncthreads();

        // ---- gather phase: lane owns ray `row`, prims k === laneHi (mod 2)
        float ax = 0.f, ay = 0.f, az = 0.f, aw = 0.f;
#pragma unroll 1
        for (int p = 0; p < 32; ++p) {
            const int k = (p << 1) | laneHi;
            const float lx = w_lds[row * 192 + k * 3 + 0];
            const float ly = w_lds[row * 192 + k * 3 + 1];
            const float lz = w_lds[row * 192 + k * 3 + 2];
            if (__builtin_fabsf(lx) < 1.f && __builtin_fabsf(ly) < 1.f &&
                __builtin_fabsf(lz) < 1.f) {
                float gx = fminf(fmaxf((lx + 1.f) * 0.5f * (MV - 1), 0.f), 14.99999f);
                float gy = fminf(fmaxf((ly + 1.f) * 0.5f * (MV - 1), 0.f), 14.99999f);
                float gz = fminf(fmaxf((lz + 1.f) * 0.5f * (MV - 1), 0.f), 14.99999f);
                int x0 = (int)gx, y0 = (int)gy, z0 = (int)gz;
                float fx = gx - x0, fy = gy - y0, fz = gz - z0;
                float wx[2] = {1.f - fx, fx};
                float wy[2] = {1.f - fy, fy};
                float wz[2] = {1.f - fz, fz};
                const int vbase = z0 * 256 + y0 * 16 + x0;
#pragma unroll
                for (int dzi = 0; dzi < 2; ++dzi)
#pragma unroll
                    for (int dyi = 0; dyi < 2; ++dyi)
#pragma unroll
                        for (int dxi = 0; dxi < 2; ++dxi) {
                            const float w = wz[dzi] * wy[dyi] * wx[dxi];
                            const int vi = vbase + dzi * 256 + dyi * 16 + dxi;
                            if (tex4) {
                                float4 c = tex4[k * 4096 + vi];   // one B128 load
                                ax += c.x * w; ay += c.y * w;
                                az += c.z * w; aw += c.w * w;
                            } else {
                                const float* pr = primrgba + k * 16384 + vi;
                                ax += pr[0 * 4096] * w; ay += pr[1 * 4096] * w;
                                az += pr[2 * 4096] * w; aw += pr[3 * 4096] * w;
                            }
                        }
            }
        }
        __syncthreads();

        // fold the two prim-halves (lane L <-> L+16 hold the same ray)
        ax += __shfl_xor(ax, 16, 32);
        ay += __shfl_xor(ay, 16, 32);
        az += __shfl_xor(az, 16, 32);
        aw += __shfl_xor(aw, 16, 32);

        float contrib = fminf(1.f, alpha + aw * DT) - alpha;
        if (!(t < tmax)) contrib = 0.f;
        rgbx += ax * contrib;
        rgby += ay * contrib;
        rgbz += az * contrib;
        alpha += contrib;
    }

    if (laneHi == 0) {
        // out layout: rayrgb [3][H][W], rayalpha [1][H][W], rayrgba [4][H][W]
        out[0 * NRAYS + rr] = rgbx;
        out[1 * NRAYS + rr] = rgby;
        out[2 * NRAYS + rr] = rgbz;
        out[3 * NRAYS + rr] = alpha;
        out[4 * NRAYS + rr] = rgbx;
        out[5 * NRAYS + rr] = rgby;
        out[6 * NRAYS + rr] = rgbz;
        out[7 * NRAYS + rr] = alpha;
    }
}

// ------------------------------- launcher ----------------------------------
extern "C" void kernel_launch(void* const* d_in, const int* in_sizes, int n_in,
                              void* d_out, int out_size, void* d_ws, size_t ws_size,
                              hipStream_t stream) {
    const float* raypos    = (const float*)d_in[0];
    const float* raydir    = (const float*)d_in[1];
    const float* tminmax   = (const float*)d_in[2];
    const float* primpos   = (const float*)d_in[3];
    const float* primrot   = (const float*)d_in[4];
    const float* primscale = (const float*)d_in[5];
    const float* primrgba  = (const float*)d_in[6];
    float* out = (float*)d_out;

    const size_t tex_bytes = (size_t)KPRIM * 4096 * sizeof(float4);  // 4 MB
    float4* tex4 = nullptr;
    if (ws_size >= tex_bytes) {
        tex4 = (float4*)d_ws;
        repack_kernel<<<(KPRIM * 4096 + 255) / 256, 256, 0, stream>>>(primrgba, tex4);
    }

    // 65536 rays / (16 rays * 4 waves per block) = 1024 blocks
    march_kernel<<<NRAYS / 64, 128, 0, stream>>>(
        raypos, raydir, tminmax, primpos, primrot, primscale, primrgba, tex4, out);
}